// RulePolicy_49744311222809
// MI455X (gfx1250) — compile-verified
//
#include <hip/hip_runtime.h>
#include <math.h>

// CDNA5 / gfx1250: wave32, WMMA f32 16x16x4.
typedef float v2f __attribute__((ext_vector_type(2)));
typedef float v8f __attribute__((ext_vector_type(8)));

#define WAVES_PER_BLOCK 8
#define BLOCK_THREADS (WAVES_PER_BLOCK * 32)

// Pairwise quantities as rank<=4 bilinear forms computed on the WMMA pipe:
//   ndist[i,j] = x_i*cj + y_i*sj - (x_j*cj + y_j*sj)                (K=3)
//   dr2[i,j]   = r2_i + r2_j - 2 x_i x_j - 2 y_i y_j                (K=4)
//   headcos    = c_i*c_j + s_i*s_j                                  (K=2)
//   delv       = v_i  (B's -v_j at K1 meets a zero in A)
// C/D tile layout (32-bit, 16x16): lane l<16 -> (M=r, N=l) in VGPR r,
// lane l>=16 -> (M=r+8, N=l-16). So each lane owns column j=(l&15) and
// 8 rows; min/argmin/any reductions stay in-lane across the i-tile loop.
__global__ __launch_bounds__(BLOCK_THREADS) void rule_policy_wmma(
    const float* __restrict__ state,
    const float* __restrict__ lengths,
    const float* __restrict__ v0p,
    const float* __restrict__ s0p,
    const float* __restrict__ dthp,
    const float* __restrict__ amaxp,
    const float* __restrict__ bpar,
    float* __restrict__ out,
    int NV)
{
  extern __shared__ __align__(16) char smem[];
  float4* P = reinterpret_cast<float4*>(smem);                    // {x, y, cos psi, sin psi}
  float2* Q = reinterpret_cast<float2*>(smem + (size_t)NV * 16);  // {v, x^2+y^2}

  const int b = blockIdx.x;
  const float* sb = state + (size_t)b * NV * 5;

  // Cooperative LDS fill (24 KB for NV=1024; WGP has 320 KB).
  for (int i = threadIdx.x; i < NV; i += BLOCK_THREADS) {
    float x = sb[5 * i + 0];
    float y = sb[5 * i + 1];
    float v = sb[5 * i + 2];
    float psi = sb[5 * i + 3];
    float sn, cs;
    __sincosf(psi, &sn, &cs);
    P[i] = make_float4(x, y, cs, sn);
    Q[i] = make_float2(v, fmaf(x, x, y * y));
  }
  __syncthreads();

  const int lane = threadIdx.x & 31;
  const int wave = threadIdx.x >> 5;
  const int jt = blockIdx.y * WAVES_PER_BLOCK + wave;  // 16-wide j tile
  const int jcol = (jt << 4) + (lane & 15);
  const int hi = lane >> 4;  // 0: lanes hold K0/K1 rows, 1: K2/K3 rows

  const float4 pj = P[jcol];
  const float2 qj = Q[jcol];
  const float xj = pj.x, yj = pj.y, cj = pj.z, sj = pj.w;
  const float vj = qj.x, r2j = qj.y;

  // B fragments (4x16 f32, 2 VGPRs): lanes 0-15 rows K0/K1, lanes 16-31 rows K2/K3.
  v2f bN, bD, bH, bV;
  bN.x = hi ? -fmaf(xj, cj, yj * sj) : cj;  bN.y = hi ? 0.f : sj;
  bD.x = hi ? 1.f : -2.f * xj;              bD.y = hi ? r2j : -2.f * yj;
  bH.x = hi ? 0.f : cj;                     bH.y = hi ? 0.f : sj;
  bV.x = hi ? 0.f : 1.f;                    bV.y = hi ? 0.f : -vj;

  const float C20SQ = 0.8830222216f;   // cos^2(20 deg)
  const float COS45 = 0.7071067812f;   // cos(45 deg)

  float best = INFINITY;
  int bestIdx = 0;
  int stop = 0;
  const int halfoff = hi << 3;
  const int ntiles = NV >> 4;

  for (int it = 0; it < ntiles; ++it) {
    const int il = (it << 4) + (lane & 15);
    const float4 pi = P[il];  // broadcast pair (lane l and l+16 read same addr)
    const float2 qi = Q[il];

    // A fragments (16x4 f32): lanes 0-15 -> {K0,K1}, lanes 16-31 -> {K2,K3}.
    v2f aN, aD, aH, aV;
    aN.x = hi ? 1.f : pi.x;   aN.y = hi ? 0.f : pi.y;
    aD.x = hi ? qi.y : pi.x;  aD.y = hi ? 1.f : pi.y;
    aH.x = hi ? 0.f : pi.z;   aH.y = hi ? 0.f : pi.w;
    aV.x = hi ? 0.f : qi.x;   aV.y = hi ? 1.f : 0.f;

    v8f z = {};
    v8f cN = __builtin_amdgcn_wmma_f32_16x16x4_f32(false, aN, false, bN, (short)0, z, false, false);
    v8f cD = __builtin_amdgcn_wmma_f32_16x16x4_f32(false, aD, false, bD, (short)0, z, false, false);
    v8f cH = __builtin_amdgcn_wmma_f32_16x16x4_f32(false, aH, false, bH, (short)0, z, false, false);
    v8f cV = __builtin_amdgcn_wmma_f32_16x16x4_f32(false, aV, false, bV, (short)0, z, false, false);

    const int ibase = (it << 4) + halfoff;
#pragma unroll
    for (int r = 0; r < 8; ++r) {
      const float nd = cN[r];
      const float d2 = cD[r];
      const float hc = cH[r];
      const float dv = cV[r];
      const float nd2 = nd * nd;
      const bool pos = nd > 0.f;
      // valid = ndist>0 && |delpsi|<20deg  (cos monotone on (-pi,pi])
      const bool valid = pos && (nd2 > d2 * C20SQ);
      const float cand = valid ? nd : INFINITY;
      if (cand < best) { best = cand; bestIdx = ibase + r; }
      // stop-any: dr<20 && |delpsi|<60 && |dpsi|>45 && v_i>0
      stop |= (pos & (d2 < 400.f) & (nd2 > 0.25f * d2) & (hc < COS45) & (dv > 0.f)) ? 1 : 0;
    }
  }

  // Combine the two M-halves of each column j (lanes l and l+16).
  const float obest = __shfl_xor(best, 16);
  const int oidx = __shfl_xor(bestIdx, 16);
  const int ostop = __shfl_xor(stop, 16);
  if (obest < best || (obest == best && oidx < bestIdx)) { best = obest; bestIdx = oidx; }
  stop |= ostop;

  if (hi == 0) {
    const float v0 = v0p[0], s0 = s0p[0], dth = dthp[0];
    const float amax = amaxp[0], bb = bpar[0];
    const float Lj = lengths[jcol];

    // Leader kinematics (bestIdx==0 when no valid leader -> sal=inf path).
    const float4 pk = P[bestIdx];
    const float2 qk = Q[bestIdx];
    const float dvx = qk.x * pk.z - vj * cj;
    const float dvy = qk.x * pk.w - vj * sj;
    const float ndv = fmaf(dvx, cj, dvy * sj);  // dv*cos(vdelpsi), exact identity

    const float sal = best - Lj;
    const float sstar = s0 + vj * dth + vj * ndv / (2.f * sqrtf(amax * bb));
    const float q = vj / v0;
    const float q2 = q * q;
    const float action_free = amax * (1.f - q2 * q2);
    const float ratio = stop ? 1.f : (sstar / sal);
    const float action_int = -amax * ratio * ratio;
    float action = (isinf(sal) || isnan(sal)) ? action_free : (action_free + action_int);
    if (stop) action = action_free + action_int;

    out[(size_t)b * NV + jcol] = action;
  }
}

extern "C" void kernel_launch(void* const* d_in, const int* in_sizes, int n_in,
                              void* d_out, int out_size, void* d_ws, size_t ws_size,
                              hipStream_t stream) {
  const float* state   = (const float*)d_in[0];
  const float* lengths = (const float*)d_in[1];
  const float* v0      = (const float*)d_in[2];
  const float* s0      = (const float*)d_in[3];
  const float* dth     = (const float*)d_in[4];
  const float* amax    = (const float*)d_in[5];
  const float* bpar    = (const float*)d_in[6];

  const int NV = in_sizes[1];            // 1024
  const int B  = in_sizes[0] / (NV * 5); // 16

  dim3 grid(B, NV / (16 * WAVES_PER_BLOCK));  // (16, 8)
  size_t shmem = (size_t)NV * (sizeof(float4) + sizeof(float2));  // 24 KB

  rule_policy_wmma<<<grid, BLOCK_THREADS, shmem, stream>>>(
      state, lengths, v0, s0, dth, amax, bpar, (float*)d_out, NV);
}